// VisionTokenController_4904852652224
// MI455X (gfx1250) — compile-verified
//
#include <hip/hip_runtime.h>

typedef __attribute__((ext_vector_type(2))) float v2f;
typedef __attribute__((ext_vector_type(4))) float v4f;
typedef __attribute__((ext_vector_type(8))) float v8f;

#define B_DIM 64
#define N_TOK 576
#define C_DIM 1024

// ---------------------------------------------------------------------------
// Kernel A: logits[b,n] = <vision[b,n,:], W[0:1024]> + <budget_emb[b,:], W[1024:2048]> + bias
// One wave (32 threads) per (b, group of 16 tokens). Uses V_WMMA_F32_16X16X4_F32:
//   A[m][k] = w[k] (same for all m)  -> lane L supplies w[k + 2*(L>>4) + {0,1}]
//   B[k][n] = x[token n][k]          -> lane L supplies x[L&15][k + 2*(L>>4) + {0,1}]
//   D[0][n] (VGPR0, lanes 0..15) = dot(w_chunk, x[n]) accumulated over chunks.
// Regular-temporal loads on purpose: this first pass over `vision` warms the
// 192MB L2 so kernel C's second pass can hit in-cache.
// ---------------------------------------------------------------------------
__global__ __launch_bounds__(32) void vtc_logits_wmma_kernel(
    const float* __restrict__ vision, const float* __restrict__ budget_emb,
    const float* __restrict__ W, const float* __restrict__ bias,
    float* __restrict__ logits)
{
    const int b    = blockIdx.y;
    const int g    = blockIdx.x;
    const int lane = threadIdx.x;          // 0..31
    const int tok  = g * 16 + (lane & 15); // token this lane feeds into B
    const int kh   = (lane >> 4) * 2;      // 0 (lanes 0-15) or 2 (lanes 16-31)

    const float* __restrict__ xrow = vision     + ((size_t)b * N_TOK + tok) * C_DIM;
    const float* __restrict__ brow = budget_emb + (size_t)b * C_DIM;

    v8f acc = {};

    // vision part: K = 0..1023
#pragma unroll 4
    for (int k = 0; k < C_DIM; k += 4) {
        v2f a = *(const v2f*)(W + k + kh);
        v2f x = *(const v2f*)(xrow + k + kh);
        acc = __builtin_amdgcn_wmma_f32_16x16x4_f32(
            /*neg_a=*/false, a, /*neg_b=*/false, x,
            /*c_mod=*/(short)0, acc, /*reuse_a=*/false, /*reuse_b=*/false);
    }
    // budget-embedding part: K = 1024..2047 (B-operand independent of token)
#pragma unroll 4
    for (int k = 0; k < C_DIM; k += 4) {
        v2f a = *(const v2f*)(W + C_DIM + k + kh);
        v2f x = *(const v2f*)(brow + k + kh);
        acc = __builtin_amdgcn_wmma_f32_16x16x4_f32(
            false, a, false, x, (short)0, acc, false, false);
    }

    // D[0][n] lives in VGPR0 of lanes 0..15 (n = lane)
    if (lane < 16) {
        logits[(size_t)b * N_TOK + tok] = acc[0] + bias[0];
    }
}

// ---------------------------------------------------------------------------
// Kernel B: exact top-K mask. Iterative argmax-of-softmax == select top-K[b]
// logits with ties broken by lowest index. rank(n) = #{m : l[m]>l[n] or
// (l[m]==l[n] && m<n)}; keep iff rank < K[b]. One block of 576 threads per b.
// ---------------------------------------------------------------------------
__global__ __launch_bounds__(N_TOK) void vtc_topk_mask_kernel(
    const float* __restrict__ logits, const float* __restrict__ token_budget,
    float* __restrict__ mask)
{
    __shared__ float l[N_TOK];
    const int b = blockIdx.x;
    const int n = threadIdx.x;

    l[n] = logits[(size_t)b * N_TOK + n];
    __syncthreads();

    const float ln = l[n];
    int rank = 0;
#pragma unroll 8
    for (int m = 0; m < N_TOK; ++m) {
        const float lm = l[m];
        rank += (int)((lm > ln) || ((lm == ln) && (m < n)));
    }

    // K = clip(trunc(budget * N), 1, N)  (torch .long() truncates toward zero)
    int K = (int)(token_budget[b] * (float)N_TOK);
    K = K < 1 ? 1 : (K > N_TOK ? N_TOK : K);

    mask[(size_t)b * N_TOK + n] = (rank < K) ? 1.0f : 0.0f;
}

// ---------------------------------------------------------------------------
// Kernel C: out[b,n,c] = vision[b,n,c] * mask[b,n].  Pure streaming, b128.
// Loads of `vision` are non-temporal (last use; may hit L2 lines warmed by
// kernel A without forcing retention). Stores are non-temporal so the 151MB
// of write data does not evict `vision` from the 192MB L2 mid-pass.
// ---------------------------------------------------------------------------
__global__ __launch_bounds__(256) void vtc_scale_kernel(
    const float* __restrict__ vision, const float* __restrict__ mask,
    float* __restrict__ out)
{
    const size_t i4 = (size_t)blockIdx.x * blockDim.x + threadIdx.x; // float4 idx
    const size_t bn = i4 >> 8;                                       // / (C/4)
    const float  m  = mask[bn];

    v4f v = __builtin_nontemporal_load((const v4f*)vision + i4);
    v4f r;
    r.x = v.x * m; r.y = v.y * m; r.z = v.z * m; r.w = v.w * m;
    __builtin_nontemporal_store(r, (v4f*)out + i4);
}

// ---------------------------------------------------------------------------
// Launch. d_out = [selected_features (B*N*C) | keep_mask (B*N)] as float.
// d_ws holds logits (B*N floats).
// ---------------------------------------------------------------------------
extern "C" void kernel_launch(void* const* d_in, const int* in_sizes, int n_in,
                              void* d_out, int out_size, void* d_ws, size_t ws_size,
                              hipStream_t stream) {
    (void)in_sizes; (void)n_in; (void)out_size; (void)ws_size;

    const float* vision       = (const float*)d_in[0]; // [B,N,C]
    const float* token_budget = (const float*)d_in[1]; // [B]
    const float* budget_emb   = (const float*)d_in[2]; // [B,C]
    const float* W            = (const float*)d_in[3]; // [1,2C]
    const float* bias         = (const float*)d_in[4]; // [1]

    float* out    = (float*)d_out;
    float* maskp  = out + (size_t)B_DIM * N_TOK * C_DIM; // second output, in d_out tail
    float* logits = (float*)d_ws;

    // A: logits via f32 WMMA GEMV. grid = (token groups, batch), one wave each.
    dim3 gridA(N_TOK / 16, B_DIM);
    vtc_logits_wmma_kernel<<<gridA, 32, 0, stream>>>(vision, budget_emb, W, bias, logits);

    // B: exact top-K mask, one block per batch row.
    vtc_topk_mask_kernel<<<B_DIM, N_TOK, 0, stream>>>(logits, token_budget, maskp);

    // C: masked copy of vision -> out (b128 streaming, NT hints).
    const size_t n4 = (size_t)B_DIM * N_TOK * C_DIM / 4;
    vtc_scale_kernel<<<(unsigned)((n4 + 255) / 256), 256, 0, stream>>>(vision, maskp, out);
}